// TinyMessagePassing_79250736545856
// MI455X (gfx1250) — compile-verified
//
#include <hip/hip_runtime.h>

#define TILE_T 256            // threads per block
#define EPT    4              // edges per thread per tile
#define TILE_E (TILE_T * EPT) // 1024 edges per tile

// ---- CDNA5 async global->LDS helpers (gfx1250, tracked by ASYNCcnt) ----
// All staged data is single-use stream traffic -> th:TH_LOAD_NT so the
// ~230MB stream does not evict the ~13MB reused node/accumulator set from
// the 192MB L2.

__device__ __forceinline__ unsigned ldsOff(const void* p) {
    // Generic pointer to LDS: low 32 bits are the workgroup-relative LDS byte
    // address (flat-aperture layout).
    return (unsigned)(unsigned long long)p;
}

__device__ __forceinline__ void asyncLoadB32NT(unsigned ldsAddr, unsigned voff,
                                               const void* base) {
    // GVS mode: mem = SGPR64 base + VGPR32 offset; VDST vgpr = LDS byte addr.
    asm volatile("global_load_async_to_lds_b32 %0, %1, %2 th:TH_LOAD_NT"
                 :: "v"(ldsAddr), "v"(voff), "s"(base)
                 : "memory");
}

__device__ __forceinline__ void asyncLoadB128NT(unsigned ldsAddr, unsigned voff,
                                                const void* base) {
    asm volatile("global_load_async_to_lds_b128 %0, %1, %2 th:TH_LOAD_NT"
                 :: "v"(ldsAddr), "v"(voff), "s"(base)
                 : "memory");
}

__device__ __forceinline__ void waitAsync0() {
    asm volatile("s_wait_asynccnt 0" ::: "memory");
}

// ---- Kernel 1: out[n] = w_node * nodes[n,0] + b_node ----

__global__ void tmp_init_kernel(const float* __restrict__ nodes,
                                const float* __restrict__ w_node_p,
                                const float* __restrict__ b_node_p,
                                float* __restrict__ out, int n_nodes) {
    int n = blockIdx.x * blockDim.x + threadIdx.x;
    if (n < n_nodes) {
        out[n] = (*w_node_p) * nodes[(size_t)n * 128] + (*b_node_p);
    }
}

// ---- Kernel 2: edge scatter, async-LDS double buffered ----

__global__ void tmp_edge_kernel(const float* __restrict__ nodes,
                                const float* __restrict__ edges,
                                const int* __restrict__ senders,
                                const int* __restrict__ receivers,
                                const float* __restrict__ w_node_p,
                                const float* __restrict__ w_edge_p,
                                float* __restrict__ out, int n_full_tiles) {
    __shared__ __align__(16) int   s_snd[2][TILE_E];
    __shared__ __align__(16) int   s_rcv[2][TILE_E];
    __shared__ __align__(16) float s_ev [2][TILE_E];

    const float wn  = *w_node_p;
    const float we  = *w_edge_p;
    const int   tid = threadIdx.x;

    // Stage one 1024-edge tile into buffer `buf` (pure async DMA, no VGPR data).
    auto stage = [&](int buf, int tile) {
        const unsigned e0 = (unsigned)tile * TILE_E;
        const unsigned i4 = e0 + (unsigned)tid * 4u;          // 4 ints / lane
        asyncLoadB128NT(ldsOff(&s_snd[buf][tid * 4]), i4 * 4u, senders);
        asyncLoadB128NT(ldsOff(&s_rcv[buf][tid * 4]), i4 * 4u, receivers);
#pragma unroll
        for (int k = 0; k < EPT; ++k) {
            const unsigned e = e0 + (unsigned)(tid + k * TILE_T);
            // edges row stride = 16 floats = 64 bytes; column 0 only.
            asyncLoadB32NT(ldsOff(&s_ev[buf][tid + k * TILE_T]), e * 64u, edges);
        }
    };

    int       cur    = 0;
    int       t      = blockIdx.x;
    const int stride = gridDim.x;

    if (t < n_full_tiles) stage(0, t);

    for (; t < n_full_tiles; t += stride) {
        waitAsync0();        // this wave's async LDS writes landed
        __syncthreads();     // all waves' tile data visible

        const int nt = t + stride;
        if (nt < n_full_tiles) stage(cur ^ 1, nt);   // prefetch next tile

#pragma unroll
        for (int k = 0; k < EPT; ++k) {
            const int   i   = tid + k * TILE_T;
            const int   s   = s_snd[cur][i];
            const int   r   = s_rcv[cur][i];
            const float ev  = s_ev[cur][i];
            const float msg = we * ev + wn * nodes[(size_t)s * 128];
            atomicAdd(&out[r], msg);   // result unused -> no-return f32 atomic
        }
        __syncthreads();     // all reads of buf[cur] done before it is restaged
        cur ^= 1;
    }
}

// ---- Kernel 3: remainder edges (direct path) ----

__global__ void tmp_tail_kernel(const float* __restrict__ nodes,
                                const float* __restrict__ edges,
                                const int* __restrict__ senders,
                                const int* __restrict__ receivers,
                                const float* __restrict__ w_node_p,
                                const float* __restrict__ w_edge_p,
                                float* __restrict__ out, int e0, int n_edges) {
    int e = e0 + blockIdx.x * blockDim.x + threadIdx.x;
    if (e < n_edges) {
        const float msg = (*w_edge_p) * edges[(size_t)e * 16] +
                          (*w_node_p) * nodes[(size_t)senders[e] * 128];
        atomicAdd(&out[receivers[e]], msg);
    }
}

extern "C" void kernel_launch(void* const* d_in, const int* in_sizes, int n_in,
                              void* d_out, int out_size, void* d_ws, size_t ws_size,
                              hipStream_t stream) {
    const float* nodes     = (const float*)d_in[0];
    const float* edges     = (const float*)d_in[1];
    const int*   senders   = (const int*)d_in[2];
    const int*   receivers = (const int*)d_in[3];
    const float* w_node    = (const float*)d_in[4];
    const float* w_edge    = (const float*)d_in[5];
    const float* b_node    = (const float*)d_in[6];
    float*       out       = (float*)d_out;

    const int n_nodes = in_sizes[0] / 128;  // D_NODE = 128
    const int n_edges = in_sizes[2];

    // 1) out = w_node * nodes[:,0] + b_node
    {
        int grid = (n_nodes + TILE_T - 1) / TILE_T;
        tmp_init_kernel<<<grid, TILE_T, 0, stream>>>(nodes, w_node, b_node,
                                                     out, n_nodes);
    }

    // 2) full tiles via async-LDS double-buffered scatter
    const int n_full = n_edges / TILE_E;
    if (n_full > 0) {
        int grid = n_full < 1024 ? n_full : 1024;
        tmp_edge_kernel<<<grid, TILE_T, 0, stream>>>(nodes, edges, senders,
                                                     receivers, w_node, w_edge,
                                                     out, n_full);
    }

    // 3) remainder edges
    const int e0 = n_full * TILE_E;
    if (e0 < n_edges) {
        int rem  = n_edges - e0;
        int grid = (rem + TILE_T - 1) / TILE_T;
        tmp_tail_kernel<<<grid, TILE_T, 0, stream>>>(nodes, edges, senders,
                                                     receivers, w_node, w_edge,
                                                     out, e0, n_edges);
    }
}